// EEG_GAT_76476187672729
// MI455X (gfx1250) — compile-verified
//
#include <hip/hip_runtime.h>
#include <hip/hip_bf16.h>
#include <math.h>

// ---------------- problem constants ----------------
#define N_NODES 3968          // 64*62
#define T_FULL  2000
#define T5      400
#define HEADS   4
#define GRAPHS  64
#define NPG     62            // nodes per graph
#define E_RAW   63488
#define E_TOT   (E_RAW + N_NODES)   // + self loops
#define MAX_K   400           // largest GAT input dim

typedef float v2f __attribute__((ext_vector_type(2)));
typedef float v8f __attribute__((ext_vector_type(8)));

__device__ __forceinline__ float preluf(float v, float a) { return v >= 0.f ? v : a * v; }
// order-preserving float -> uint encode for atomicMax-based segment max
__device__ __forceinline__ unsigned fenc(float f) {
    unsigned u = __float_as_uint(f);
    return (u & 0x80000000u) ? ~u : (u | 0x80000000u);
}
__device__ __forceinline__ float fdec(unsigned e) {
    unsigned u = (e & 0x80000000u) ? (e & 0x7fffffffu) : ~e;
    return __uint_as_float(u);
}

// =====================================================================
// Front end: per-(b,ch) row, conv chain recomputed per pass in LDS.
// depth = how many conv layers to evaluate this pass.
// finalMode=0: accumulate BN stats (sum, sumsq) of raw layer-`depth` output.
// finalMode=1: apply BN3+PReLU and fold agg_w into xagg.
// =====================================================================
__global__ void front_pass_kernel(
    const float* __restrict__ x,
    const float* w0, const float* b0, const float* g0, const float* be0, const float* a0,
    const float* w1, const float* b1, const float* g1, const float* be1, const float* a1,
    const float* w2, const float* b2, const float* g2, const float* be2, const float* a2,
    const float* __restrict__ meanb, const float* __restrict__ invb,
    int K, int depth, int finalMode,
    float* statsOut,
    float* __restrict__ xagg, const float* __restrict__ agg_w, int branch, int accumulate)
{
    __shared__ float smem[T_FULL + 6 * T_FULL];   // in-row + two 3-channel ping-pong buffers
    __shared__ float sred[6];
    float* sIn  = smem;
    float* bufA = smem + T_FULL;
    float* bufB = bufA + 3 * T_FULL;

    const float* Ws[3]  = { w0, w1, w2 };
    const float* Bs[3]  = { b0, b1, b2 };
    const float* Gs[3]  = { g0, g1, g2 };
    const float* Bes[3] = { be0, be1, be2 };
    const float* As[3]  = { a0, a1, a2 };

    const int row = blockIdx.x, tid = threadIdx.x, nt = blockDim.x;
    const float* xr = x + (size_t)row * T_FULL;
    for (int t = tid; t < T_FULL; t += nt) sIn[t] = xr[t];
    if (tid < 6) sred[tid] = 0.f;
    __syncthreads();

    const int pad = K >> 1;
    float lsum[3] = {0.f, 0.f, 0.f}, lsq[3] = {0.f, 0.f, 0.f};

    for (int l = 0; l < depth; l++) {
        const int   cin  = (l == 0) ? 1 : 3;
        const float* srcb = (l == 0) ? sIn : ((l == 1) ? bufA : bufB);
        float*       dstb = (l == 0) ? bufA : bufB;
        const bool  last = (l == depth - 1);
        const float* Wl = Ws[l];
        const float bb[3] = { Bs[l][0], Bs[l][1], Bs[l][2] };

        for (int t = tid; t < T_FULL; t += nt) {
            float z[3] = { bb[0], bb[1], bb[2] };
            for (int ci = 0; ci < cin; ci++) {
                const float* s = srcb + ci * T_FULL;
                for (int j = 0; j < K; j++) {
                    int tt = t + j - pad;
                    float sv = (tt >= 0 && tt < T_FULL) ? s[tt] : 0.f;
#pragma unroll
                    for (int c = 0; c < 3; c++) z[c] += Wl[(c * cin + ci) * K + j] * sv;
                }
            }
            if (!last) {
                float a = As[l][0];
#pragma unroll
                for (int c = 0; c < 3; c++) {
                    float y = Gs[l][c] * (z[c] - meanb[l * 3 + c]) * invb[l * 3 + c] + Bes[l][c];
                    dstb[c * T_FULL + t] = preluf(y, a);
                }
            } else if (finalMode) {
                float a = As[l][0];
                float acc = 0.f;
#pragma unroll
                for (int c = 0; c < 3; c++) {
                    float y = Gs[l][c] * (z[c] - meanb[l * 3 + c]) * invb[l * 3 + c] + Bes[l][c];
                    acc += preluf(y, a) * agg_w[branch * 3 + c];
                }
                size_t o = (size_t)row * T_FULL + t;
                if (accumulate) xagg[o] += acc; else xagg[o] = acc;
            } else {
#pragma unroll
                for (int c = 0; c < 3; c++) { lsum[c] += z[c]; lsq[c] += z[c] * z[c]; }
            }
        }
        __syncthreads();
    }
    if (!finalMode) {
#pragma unroll
        for (int c = 0; c < 3; c++) { atomicAdd(&sred[c], lsum[c]); atomicAdd(&sred[3 + c], lsq[c]); }
        __syncthreads();
        if (tid < 6) atomicAdd(&statsOut[tid], sred[tid]);
    }
}

__global__ void finalize_stats_kernel(const float* __restrict__ stats,
                                      float* __restrict__ mean, float* __restrict__ invstd,
                                      int nch, float invN)
{
    int c = threadIdx.x;
    if (c < nch) {
        float m = stats[c] * invN;
        float v = stats[nch + c] * invN - m * m;
        mean[c]   = m;
        invstd[c] = rsqrtf(v + 1e-5f);
    }
}

// ---------------- downsample-by-5 + its BN stats ----------------
__global__ void dc_stats_kernel(const float* __restrict__ xagg,
                                const float* __restrict__ dc_w, const float* __restrict__ dc_b,
                                const float* __restrict__ agg_b,
                                float* __restrict__ xs, float* statsOut)
{
    __shared__ float sred[2];
    int tid = threadIdx.x;
    int idx = blockIdx.x * blockDim.x + tid;
    if (tid < 2) sred[tid] = 0.f;
    __syncthreads();
    float ls = 0.f, lq = 0.f;
    if (idx < N_NODES * T5) {
        int row = idx / T5, e = idx % T5;
        const float* p = xagg + (size_t)row * T_FULL + e * 5;
        float wsum = dc_w[0] + dc_w[1] + dc_w[2] + dc_w[3] + dc_w[4];
        float s = dc_b[0] + agg_b[0] * wsum;
#pragma unroll
        for (int j = 0; j < 5; j++) s += p[j] * dc_w[j];
        xs[idx] = s;
        ls = s; lq = s * s;
    }
    atomicAdd(&sred[0], ls);
    atomicAdd(&sred[1], lq);
    __syncthreads();
    if (tid < 2) atomicAdd(&statsOut[tid], sred[tid]);
}

__global__ void dc_norm_kernel(float* __restrict__ xs,
                               const float* __restrict__ g, const float* __restrict__ be,
                               const float* __restrict__ a,
                               const float* __restrict__ mean, const float* __restrict__ invstd,
                               int tot)
{
    int i = blockIdx.x * blockDim.x + threadIdx.x;
    if (i >= tot) return;
    float v = g[0] * (xs[i] - mean[0]) * invstd[0] + be[0];
    xs[i] = preluf(v, a[0]);
}

// =====================================================================
// GAT linear transform on the matrix pipe: native f32 WMMA 16x16x4.
// One wave per 16x16 output tile.
// The B tile (K x 16 weight columns, reused by all 248 row tiles) is
// staged into LDS with the gfx1250 async LDS-copy path (ASYNCcnt), then
// the K-loop reads B from LDS (no per-iteration 64-bit address math) and
// streams A contiguously from global with prefetch.
// =====================================================================
__global__ void __launch_bounds__(32) wmma_gemm_f32_kernel(
    const float* __restrict__ A,   // n x K   (row major)
    const float* __restrict__ B,   // K x N   (row major)
    float* __restrict__ C,         // n x N
    int K, int N)
{
    __shared__ float shB[MAX_K * 16];     // B tile, row-major K x 16

    const int lane = threadIdx.x;
    const int half = lane >> 4;     // 0: K pair {0,1}, 1: K pair {2,3}
    const int l16  = lane & 15;
    const int m0 = blockIdx.x * 16;
    const int n0 = blockIdx.y * 16;

    // ---- async-stage B tile into LDS: 16B chunk c = (row k, sub 0..3) ----
    // chunk counts (K*4) are exact multiples of 32 for K in {400,128,64}.
    {
        const int chunks = K * 4;
        for (int c = lane; c < chunks; c += 32) {
            const int row = c >> 2, sub = c & 3;
            const float* gp = B + (size_t)row * N + n0 + sub * 4;
            unsigned laddr = (unsigned)(size_t)(&shB[0]) + (unsigned)c * 16u;
            asm volatile("global_load_async_to_lds_b128 %0, %1, off"
                         :: "v"(laddr), "v"(gp)
                         : "memory");
        }
        asm volatile("s_wait_asynccnt 0x0" ::: "memory");
    }

    const float* arow = A + (size_t)(m0 + l16) * K + half * 2;
    const float* bl   = shB + l16;        // column l16, rows advance by 16 floats

    v8f acc = {};
    for (int k0 = 0; k0 < K; k0 += 4) {
        if (k0 + 40 < K) __builtin_prefetch(&arow[k0 + 40], 0, 0);  // global_prefetch_b8
        v2f a, b;
        a.x = arow[k0];
        a.y = arow[k0 + 1];
        const int kb = k0 + half * 2;
        b.x = bl[kb * 16];
        b.y = bl[(kb + 1) * 16];
        acc = __builtin_amdgcn_wmma_f32_16x16x4_f32(
            /*neg_a=*/false, a, /*neg_b=*/false, b,
            /*c_mod=*/(short)0, acc, /*reuse_a=*/false, /*reuse_b=*/false);
    }
    // D layout: VGPR r -> row m0 + half*8 + r, col n0 + l16
    float* crow = C + (size_t)(m0 + half * 8) * N + n0 + l16;
#pragma unroll
    for (int r = 0; r < 8; r++) crow[(size_t)r * N] = acc[r];
}

// ---------------- attention coefficients ----------------
__global__ void attn_coeff_kernel(const float* __restrict__ hw,
                                  const float* __restrict__ att_src, const float* __restrict__ att_dst,
                                  float* __restrict__ asb, float* __restrict__ adb, int Ff)
{
    int i = blockIdx.x * blockDim.x + threadIdx.x;
    if (i >= N_NODES * HEADS) return;
    int node = i / HEADS, h = i % HEADS;
    const float* hp = hw + (size_t)node * HEADS * Ff + h * Ff;
    float s = 0.f, d = 0.f;
    for (int f = 0; f < Ff; f++) {
        s += hp[f] * att_src[h * Ff + f];
        d += hp[f] * att_dst[h * Ff + f];
    }
    asb[i] = s;
    adb[i] = d;
}

// ---------------- 3-pass edge softmax + aggregation ----------------
__global__ void edge_logits_kernel(const int* __restrict__ ei,
                                   const float* __restrict__ asb, const float* __restrict__ adb,
                                   float* __restrict__ ebuf, unsigned* __restrict__ mkey)
{
    int e = blockIdx.x * blockDim.x + threadIdx.x;
    if (e >= E_TOT) return;
    int src = (e < E_RAW) ? ei[e]          : (e - E_RAW);
    int dst = (e < E_RAW) ? ei[E_RAW + e]  : (e - E_RAW);
#pragma unroll
    for (int h = 0; h < HEADS; h++) {
        float v = asb[src * HEADS + h] + adb[dst * HEADS + h];
        v = (v > 0.f) ? v : 0.2f * v;              // leaky_relu(0.2)
        ebuf[(size_t)e * HEADS + h] = v;
        atomicMax(&mkey[dst * HEADS + h], fenc(v)); // segment max
    }
}

__global__ void edge_exp_kernel(const int* __restrict__ ei,
                                float* __restrict__ ebuf, const unsigned* __restrict__ mkey,
                                float* __restrict__ den)
{
    int e = blockIdx.x * blockDim.x + threadIdx.x;
    if (e >= E_TOT) return;
    int dst = (e < E_RAW) ? ei[E_RAW + e] : (e - E_RAW);
#pragma unroll
    for (int h = 0; h < HEADS; h++) {
        float m  = fdec(mkey[dst * HEADS + h]);
        float ex = expf(ebuf[(size_t)e * HEADS + h] - m);
        ebuf[(size_t)e * HEADS + h] = ex;
        atomicAdd(&den[dst * HEADS + h], ex);
    }
}

__global__ void edge_agg_kernel(const int* __restrict__ ei,
                                const float* __restrict__ ebuf, const float* __restrict__ den,
                                const float* __restrict__ hw, float* __restrict__ outacc, int Ff)
{
    int idx = blockIdx.x * blockDim.x + threadIdx.x;
    if (idx >= E_TOT * HEADS) return;
    int e = idx >> 2, h = idx & 3;
    int src = (e < E_RAW) ? ei[e]         : (e - E_RAW);
    int dst = (e < E_RAW) ? ei[E_RAW + e] : (e - E_RAW);
    float alpha = ebuf[idx] / (den[dst * HEADS + h] + 1e-16f);
    const float* hs = hw     + (size_t)src * HEADS * Ff + h * Ff;
    float*       o  = outacc + (size_t)dst * HEADS * Ff + h * Ff;
    for (int f = 0; f < Ff; f++) atomicAdd(&o[f], alpha * hs[f]);
}

// ---------------- GAT bias + GraphNorm + PReLU ----------------
__global__ void graphnorm_kernel(const float* __restrict__ outacc, const float* __restrict__ gbias,
                                 const float* __restrict__ w, const float* __restrict__ b,
                                 const float* __restrict__ ms, const float* __restrict__ a,
                                 float* __restrict__ hout, int D)
{
    int g = blockIdx.x, d = threadIdx.x;
    if (d >= D) return;
    const float* base = outacc + (size_t)g * NPG * D + d;
    float gb = gbias[d];
    float s = 0.f;
    for (int i = 0; i < NPG; i++) s += base[(size_t)i * D] + gb;
    float sub = ms[d] * (s / (float)NPG);
    float v = 0.f;
    for (int i = 0; i < NPG; i++) { float o = base[(size_t)i * D] + gb - sub; v += o * o; }
    float rs = rsqrtf(v / (float)NPG + 1e-5f);
    float aa = a[0], wd = w[d], bd = b[d];
    for (int i = 0; i < NPG; i++) {
        float o = (base[(size_t)i * D] + gb - sub) * rs;
        o = wd * o + bd;
        hout[((size_t)g * NPG + i) * D + d] = preluf(o, aa);
    }
}

// ---------------- mean pooling + MLP(ELU) + softmax ----------------
__global__ void pool_mlp_kernel(const float* __restrict__ h,
                                const float* w1, const float* b1,
                                const float* w2, const float* b2,
                                const float* w3, const float* b3,
                                float* __restrict__ out)
{
    __shared__ float emb[32], z1[16], z2[8], z3[4];
    int g = blockIdx.x, t = threadIdx.x;
    float s = 0.f;
    for (int i = 0; i < NPG; i++) s += h[((size_t)g * NPG + i) * 32 + t];
    emb[t] = s / (float)NPG;
    __syncthreads();
    if (t < 16) { float z = b1[t]; for (int i = 0; i < 32; i++) z += emb[i] * w1[i * 16 + t]; z1[t] = z > 0.f ? z : expf(z) - 1.f; }
    __syncthreads();
    if (t < 8)  { float z = b2[t]; for (int i = 0; i < 16; i++) z += z1[i] * w2[i * 8 + t];  z2[t] = z > 0.f ? z : expf(z) - 1.f; }
    __syncthreads();
    if (t < 4)  { float z = b3[t]; for (int i = 0; i < 8;  i++) z += z2[i] * w3[i * 4 + t];  z3[t] = z > 0.f ? z : expf(z) - 1.f; }
    __syncthreads();
    if (t == 0) {
        float mx = z3[0];
        for (int j = 1; j < 4; j++) mx = fmaxf(mx, z3[j]);
        float e[4], se = 0.f;
        for (int j = 0; j < 4; j++) { e[j] = expf(z3[j] - mx); se += e[j]; }
        for (int j = 0; j < 4; j++) out[(size_t)g * 4 + j] = e[j] / se;
    }
}

// =====================================================================
// Host orchestration
// =====================================================================
extern "C" void kernel_launch(void* const* d_in, const int* in_sizes, int n_in,
                              void* d_out, int out_size, void* d_ws, size_t ws_size,
                              hipStream_t stream)
{
    (void)in_sizes; (void)n_in; (void)out_size; (void)ws_size;
    auto F = [&](int i) { return (const float*)d_in[i]; };

    // --- flattened input indices (dict insertion order, depth-first) ---
    const float* x     = F(0);
    const float* agg_w = F(1);
    const float* agg_b = F(2);
    const float* dc_w  = F(3);
    const float* dc_b  = F(4);
    const float* dc_g  = F(5);
    const float* dc_be = F(6);
    const float* dc_a  = F(7);
    // 8..52: branch0/1/2, each 3 layers x (w,b,gamma,beta,a)
    // 53..68: gats (w, att_src, att_dst, bias) x4
    // 69..84: gnorms (weight, bias, mean_scale, a) x4
    // 85..90: mlp w1,b1,w2,b2,w3,b3
    const int* ei = (const int*)d_in[91];   // (2, 63488)
    // d_in[92] = batch (contiguous repeat(arange(64),62)) -- structure hardcoded

    // --- workspace carve ---
    size_t off = 0;
    auto carve = [&](size_t bytes) -> char* {
        char* p = (char*)d_ws + off;
        off += (bytes + 255) & ~(size_t)255;
        return p;
    };
    float*    xagg   = (float*)carve((size_t)N_NODES * T_FULL * 4);   // 31.7 MB
    float*    H0     = (float*)carve((size_t)N_NODES * T5 * 4);       // node features (n x 400)
    float*    hw     = (float*)carve((size_t)N_NODES * 128 * 4);
    float*    h1     = (float*)carve((size_t)N_NODES * 128 * 4);
    float*    h2     = (float*)carve((size_t)N_NODES * 128 * 4);
    float*    outacc = (float*)carve((size_t)N_NODES * 128 * 4);
    float*    asb    = (float*)carve((size_t)N_NODES * HEADS * 4);
    float*    adb    = (float*)carve((size_t)N_NODES * HEADS * 4);
    unsigned* mkey   = (unsigned*)carve((size_t)N_NODES * HEADS * 4);
    float*    den    = (float*)carve((size_t)N_NODES * HEADS * 4);
    float*    ebuf   = (float*)carve((size_t)E_TOT * HEADS * 4);
    float*    stats  = (float*)carve(64);
    float*    meanb  = (float*)carve(64);
    float*    invb   = (float*)carve(64);
    float*    dcm    = (float*)carve(64);
    float*    dci    = (float*)carve(64);

    // --- front end: 3 branches, streaming stats passes then final aggregate ---
    const int Ks[3] = {15, 7, 3};
    const float invN_front = 1.f / (float)((size_t)N_NODES * T_FULL);
    for (int br = 0; br < 3; br++) {
        int base = 8 + br * 15;
        const float* P[15];
        for (int i = 0; i < 15; i++) P[i] = F(base + i);
        int K = Ks[br];
        for (int depth = 1; depth <= 3; depth++) {
            hipMemsetAsync(stats, 0, 6 * sizeof(float), stream);
            front_pass_kernel<<<N_NODES, 256, 0, stream>>>(x,
                P[0], P[1], P[2], P[3], P[4],
                P[5], P[6], P[7], P[8], P[9],
                P[10], P[11], P[12], P[13], P[14],
                meanb, invb, K, depth, 0, stats, nullptr, agg_w, br, 0);
            finalize_stats_kernel<<<1, 3, 0, stream>>>(stats, meanb + 3 * (depth - 1),
                                                       invb + 3 * (depth - 1), 3, invN_front);
        }
        front_pass_kernel<<<N_NODES, 256, 0, stream>>>(x,
            P[0], P[1], P[2], P[3], P[4],
            P[5], P[6], P[7], P[8], P[9],
            P[10], P[11], P[12], P[13], P[14],
            meanb, invb, K, 3, 1, nullptr, xagg, agg_w, br, br > 0 ? 1 : 0);
    }

    // --- downsample by 5 + BN + PReLU -> H0 (n x 400) ---
    const int totdc = N_NODES * T5;
    hipMemsetAsync(stats, 0, 2 * sizeof(float), stream);
    dc_stats_kernel<<<(totdc + 255) / 256, 256, 0, stream>>>(xagg, dc_w, dc_b, agg_b, H0, stats);
    finalize_stats_kernel<<<1, 1, 0, stream>>>(stats, dcm, dci, 1, 1.f / (float)totdc);
    dc_norm_kernel<<<(totdc + 255) / 256, 256, 0, stream>>>(H0, dc_g, dc_be, dc_a, dcm, dci, totdc);

    // --- 4 GAT layers ---
    const int cins[4] = {400, 128, 128, 64};
    const int HFs[4]  = {128, 128, 64, 32};
    const int Fs[4]   = {32, 32, 16, 8};
    const float* hin = H0;
    float* houts[2] = { h1, h2 };
    for (int l = 0; l < 4; l++) {
        int gb = 53 + l * 4;
        const float* gw    = F(gb);
        const float* gas   = F(gb + 1);
        const float* gad   = F(gb + 2);
        const float* gbias = F(gb + 3);
        int nb = 69 + l * 4;
        const float* nw  = F(nb);
        const float* nbi = F(nb + 1);
        const float* nms = F(nb + 2);
        const float* na  = F(nb + 3);
        int cin = cins[l], HF = HFs[l], Ff = Fs[l];

        dim3 gg(N_NODES / 16, HF / 16);
        wmma_gemm_f32_kernel<<<gg, 32, 0, stream>>>(hin, gw, hw, cin, HF);

        int nh = N_NODES * HEADS;
        attn_coeff_kernel<<<(nh + 255) / 256, 256, 0, stream>>>(hw, gas, gad, asb, adb, Ff);

        hipMemsetAsync(mkey,   0, (size_t)nh * 4, stream);
        hipMemsetAsync(den,    0, (size_t)nh * 4, stream);
        hipMemsetAsync(outacc, 0, (size_t)N_NODES * HF * 4, stream);

        edge_logits_kernel<<<(E_TOT + 255) / 256, 256, 0, stream>>>(ei, asb, adb, ebuf, mkey);
        edge_exp_kernel<<<(E_TOT + 255) / 256, 256, 0, stream>>>(ei, ebuf, mkey, den);
        edge_agg_kernel<<<(E_TOT * HEADS + 255) / 256, 256, 0, stream>>>(ei, ebuf, den, hw, outacc, Ff);

        float* hout = houts[l & 1];
        graphnorm_kernel<<<GRAPHS, HF, 0, stream>>>(outacc, gbias, nw, nbi, nms, na, hout, HF);
        hin = hout;
    }

    // --- pooling + MLP + softmax -> d_out (64 x 4 f32) ---
    pool_mlp_kernel<<<GRAPHS, 32, 0, stream>>>(hin, F(85), F(86), F(87), F(88), F(89), F(90),
                                               (float*)d_out);
}